// PointTransformerTransitionUp_5617817224084
// MI455X (gfx1250) — compile-verified
//
#include <hip/hip_runtime.h>
#include <hip/hip_bf16.h>

#define BATCH   2
#define NPTS    16384      // N high points
#define SPTS    4096       // S low points
#define CLOW    512
#define CHIGH   256
#define BN_EPS  1e-5f
#define NN_EPS  1e-8f

typedef float v2f __attribute__((ext_vector_type(2)));
typedef float v8f __attribute__((ext_vector_type(8)));

// ---------------------------------------------------------------- K0: zero stats
__global__ void k0_zero(float* __restrict__ stats) {
    int i = blockIdx.x * 256 + threadIdx.x;   // 512 floats: sum + sumsq
    if (i < 512) stats[i] = 0.0f;
}

// ---------------------------------------------------------------- K1: WMMA GEMM + bias + stats
// hpre[b, s, o] = sum_c W[o,c] * points_low[b,c,s] + bias[o]
// grid: (S/128, CHIGH/16, B), block 256 (8 waves, each a 16s x 16o tile)
__global__ __launch_bounds__(256)
void k1_gemm(const float* __restrict__ PL, const float* __restrict__ W,
             const float* __restrict__ bias, float* __restrict__ hpre,
             float* __restrict__ gsum, float* __restrict__ gsumsq)
{
    __shared__ float Wl[CLOW * 17];   // [c][o_local], stride 17 (conflict-free)
    __shared__ float red[32];         // 16 sums + 16 sumsqs

    const int b   = blockIdx.z;
    const int o0  = blockIdx.y * 16;
    const int sbl = blockIdx.x * 128;
    const int tid = threadIdx.x;

    // stage W[o0..o0+15][0..511] -> LDS, coalesced in c
    for (int i = tid; i < 16 * CLOW; i += 256) {
        int ol = i / CLOW, c = i % CLOW;
        Wl[c * 17 + ol] = W[(o0 + ol) * CLOW + c];
    }
    if (tid < 32) red[tid] = 0.0f;
    __syncthreads();

    const int wave  = tid >> 5;
    const int lane  = tid & 31;
    const int srow  = lane & 15;      // M / N sub-index
    const int khalf = lane >> 4;      // 0 or 1: which K pair
    const int s0    = sbl + wave * 16;

    const float* plb = PL + (size_t)b * CLOW * SPTS;

    v8f acc = {};
    for (int c0 = 0; c0 < CLOW; c0 += 4) {
        const int cA = c0 + khalf * 2;
        v2f a, bm;
        // A (16x4, M=s, K=c): lane<16 -> K = c0,c0+1 ; lane>=16 -> K = c0+2,c0+3
        a.x = plb[(size_t)cA       * SPTS + s0 + srow];
        a.y = plb[(size_t)(cA + 1) * SPTS + s0 + srow];
        // B (4x16, K=c, N=o): same K split per lane-half, column = srow
        bm.x = Wl[cA       * 17 + srow];
        bm.y = Wl[(cA + 1) * 17 + srow];
        acc = __builtin_amdgcn_wmma_f32_16x16x4_f32(
            false, a, false, bm, (short)0, acc, false, false);
    }

    // bias + store hpre[b, s, o] (o contiguous) + stats
    const float bo = bias[o0 + srow];
    float psum = 0.0f, psq = 0.0f;
    float* hb = hpre + (size_t)b * SPTS * CHIGH;
#pragma unroll
    for (int r = 0; r < 8; r++) {
        float v = acc[r] + bo;
        int   s = s0 + r + khalf * 8;          // M = r (+8 for upper lane half)
        hb[(size_t)s * CHIGH + o0 + srow] = v; // lanes 0-15 / 16-31 each 64B contiguous
        psum += v;
        psq  += v * v;
    }
    // lanes l and l+16 hold the same output channel -> combine, then LDS, then global
    psum += __shfl_xor(psum, 16);
    psq  += __shfl_xor(psq, 16);
    if (lane < 16) {
        atomicAdd(&red[srow],      psum);
        atomicAdd(&red[16 + srow], psq);
    }
    __syncthreads();
    if (tid < 16) {
        atomicAdd(&gsum[o0 + tid],   red[tid]);
        atomicAdd(&gsumsq[o0 + tid], red[16 + tid]);
    }
}

// ---------------------------------------------------------------- K2: batchnorm scale/shift
__global__ void k2_scale(const float* __restrict__ gsum, const float* __restrict__ gsumsq,
                         const float* __restrict__ gamma, const float* __restrict__ beta,
                         float* __restrict__ scale, float* __restrict__ shift)
{
    int c = threadIdx.x;
    const float inv_n = 1.0f / (float)(BATCH * SPTS);
    float m  = gsum[c] * inv_n;
    float v  = gsumsq[c] * inv_n - m * m;
    float sc = gamma[c] * rsqrtf(v + BN_EPS);
    scale[c] = sc;
    shift[c] = beta[c] - m * sc;
}

// ---------------------------------------------------------------- K3: normalize + ReLU in place
__global__ __launch_bounds__(256)
void k3_norm(float* __restrict__ hpre, const float* __restrict__ scale,
             const float* __restrict__ shift)
{
    size_t i = (size_t)blockIdx.x * 256 + threadIdx.x;   // exactly B*S*CHIGH threads
    int c = (int)(i & (CHIGH - 1));
    float v = fmaf(hpre[i], scale[c], shift[c]);
    hpre[i] = fmaxf(v, 0.0f);
}

// ---------------------------------------------------------------- K4: three_nn (top-3 smallest d2)
// grid: (N/256, B), block 256.
// (x, y, z, |p|^2) packed as float4 in LDS (64 KB) -> one ds_load_b128 per
// candidate (broadcast read, all lanes same address). Inner loop:
// 1 ds_b128 + 3 FMA + 1 add + top-3 select chain.
__global__ __launch_bounds__(256)
void k4_nn(const float* __restrict__ xyz_low, const float* __restrict__ xyz_high,
           int* __restrict__ oidx, float* __restrict__ owgt)
{
    __shared__ float4 pl[SPTS];   // 64 KB, well within the 320 KB/WGP budget

    const int b = blockIdx.y;
    const int n = blockIdx.x * 256 + threadIdx.x;

    const float* xl = xyz_low + (size_t)b * 3 * SPTS;
    for (int i = threadIdx.x; i < SPTS; i += 256) {
        float x = xl[i], y = xl[SPTS + i], z = xl[2 * SPTS + i];
        pl[i] = make_float4(x, y, z, x * x + y * y + z * z);
    }
    __syncthreads();

    const float* xh = xyz_high + (size_t)b * 3 * NPTS;
    const float hx = xh[n], hy = xh[NPTS + n], hz = xh[2 * NPTS + n];
    const float h2  = hx * hx + hy * hy + hz * hz;
    const float nhx = -2.0f * hx, nhy = -2.0f * hy, nhz = -2.0f * hz;

    float d0 = 3.4e38f, d1 = 3.4e38f, d2 = 3.4e38f;
    int   i0 = 0, i1 = 0, i2 = 0;

#pragma unroll 4
    for (int s = 0; s < SPTS; s++) {
        float4 p = pl[s];
        float  d = fmaf(nhx, p.x, fmaf(nhy, p.y, fmaf(nhz, p.z, h2))) + p.w;
        if (d < d2) {
            if (d < d1) {
                d2 = d1; i2 = i1;
                if (d < d0) { d1 = d0; i1 = i0; d0 = d; i0 = s; }
                else        { d1 = d;  i1 = s; }
            } else { d2 = d; i2 = s; }
        }
    }

    d0 = fmaxf(d0, 0.0f); d1 = fmaxf(d1, 0.0f); d2 = fmaxf(d2, 0.0f);
    float r0 = 1.0f / (d0 + NN_EPS);
    float r1 = 1.0f / (d1 + NN_EPS);
    float r2 = 1.0f / (d2 + NN_EPS);
    float rs = 1.0f / (r0 + r1 + r2);

    size_t base = ((size_t)b * NPTS + n) * 3;
    oidx[base] = i0; oidx[base + 1] = i1; oidx[base + 2] = i2;
    owgt[base] = r0 * rs; owgt[base + 1] = r1 * rs; owgt[base + 2] = r2 * rs;
}

// ---------------------------------------------------------------- K5: gather + weighted sum + residual
// grid: (N/32, B), block 256. Coalesced gather over channels (lane = channel ->
// contiguous 1KB reads per target); LDS transpose (odd stride 261, conflict-free
// in both phases) for n-contiguous output writes; residual added on the way out.
__global__ __launch_bounds__(256)
void k5_interp(const float* __restrict__ hpre, const int* __restrict__ oidx,
               const float* __restrict__ owgt, const float* __restrict__ phigh,
               float* __restrict__ out)
{
    __shared__ float accl[32 * 261];
    __shared__ int   sidx[96];
    __shared__ float swgt[96];

    const int b   = blockIdx.y;
    const int n0  = blockIdx.x * 32;
    const int tid = threadIdx.x;

    size_t base = ((size_t)b * NPTS + n0) * 3;
    if (tid < 96) { sidx[tid] = oidx[base + tid]; swgt[tid] = owgt[base + tid]; }
    __syncthreads();

    const float* hb = hpre + (size_t)b * SPTS * CHIGH;
    const int c = tid;  // 256 channels == 256 threads
    for (int nn = 0; nn < 32; nn++) {
        float a = 0.0f;
#pragma unroll
        for (int k = 0; k < 3; k++) {
            int   s = sidx[nn * 3 + k];
            float w = swgt[nn * 3 + k];
            a = fmaf(w, hb[(size_t)s * CHIGH + c], a);
        }
        accl[nn * 261 + c] = a;
    }
    __syncthreads();

    const float* phb = phigh + (size_t)b * CHIGH * NPTS;
    float*       ob  = out   + (size_t)b * CHIGH * NPTS;
    const int nl   = tid & 31;   // consecutive lanes -> consecutive n (coalesced)
    const int crow = tid >> 5;   // 8 channels per pass
    for (int cp = 0; cp < 32; cp++) {
        int cc = cp * 8 + crow;
        size_t o = (size_t)cc * NPTS + n0 + nl;
        ob[o] = accl[nl * 261 + cc] + phb[o];
    }
}

// ---------------------------------------------------------------- launch
extern "C" void kernel_launch(void* const* d_in, const int* in_sizes, int n_in,
                              void* d_out, int out_size, void* d_ws, size_t ws_size,
                              hipStream_t stream) {
    const float* xyz_low     = (const float*)d_in[0];
    const float* xyz_high    = (const float*)d_in[1];
    const float* points_low  = (const float*)d_in[2];
    const float* points_high = (const float*)d_in[3];
    const float* W           = (const float*)d_in[4];
    const float* bias        = (const float*)d_in[5];
    const float* gamma       = (const float*)d_in[6];
    const float* beta        = (const float*)d_in[7];

    char* ws = (char*)d_ws;
    const size_t hpre_bytes = (size_t)BATCH * SPTS * CHIGH * sizeof(float); // 8 MB
    float* hpre  = (float*)ws;
    float* gsum  = (float*)(ws + hpre_bytes);          // 256
    float* gsq   = gsum + 256;                         // 256
    float* scale = gsum + 512;                         // 256
    float* shift = gsum + 768;                         // 256
    int*   idx   = (int*)(ws + hpre_bytes + 4096);     // B*N*3 ints
    float* wgt   = (float*)(ws + hpre_bytes + 4096 + (size_t)BATCH * NPTS * 3 * sizeof(int));

    k0_zero<<<2, 256, 0, stream>>>(gsum);

    dim3 g1(SPTS / 128, CHIGH / 16, BATCH);
    k1_gemm<<<g1, 256, 0, stream>>>(points_low, W, bias, hpre, gsum, gsq);

    k2_scale<<<1, CHIGH, 0, stream>>>(gsum, gsq, gamma, beta, scale, shift);

    k3_norm<<<(BATCH * SPTS * CHIGH) / 256, 256, 0, stream>>>(hpre, scale, shift);

    dim3 g4(NPTS / 256, BATCH);
    k4_nn<<<g4, 256, 0, stream>>>(xyz_low, xyz_high, idx, wgt);

    dim3 g5(NPTS / 32, BATCH);
    k5_interp<<<g5, 256, 0, stream>>>(hpre, idx, wgt, points_high, (float*)d_out);
}